// BatchCRF_77343771067069
// MI455X (gfx1250) — compile-verified
//
#include <hip/hip_runtime.h>
#include <stdint.h>

// Viterbi decode (max-plus semiring), N=64 chains, C=8 tags, L=32768 steps.
// Forward: 8 lanes per chain (lane = next tag), ds_swizzle butterfly exchange,
//          async double-buffered LDS staging of the observation streams.
// Backward: 1 lane per chain, packed 3-bit backpointers (1 u32 per (t,n)).

#define N_SEQ   64
#define C_TAG   8
#define L_LEN   32768
#define CHUNK   32                    // time steps staged per refill
#define NCHUNK  (L_LEN / CHUNK)
#define FWD_BLK 32                    // one wave32 -> 4 sequences per block
#define SEQ_PB  (FWD_BLK / C_TAG)

__device__ __forceinline__ uint32_t lds_addr32(const void* p) {
  // Generic pointers into LDS carry the workgroup-relative byte offset in
  // their low 32 bits (CDNA5 ISA 10.2 aperture layout).
  return (uint32_t)(uintptr_t)p;
}

// Issue one CHUNK*4-byte (128B) async refill: 8 x b128, per-lane LDS slot.
// Note: INST_OFFSET is added to BOTH the LDS and the global address
// (ISA 15.18 async pseudocode), so one immediate advances both streams.
__device__ __forceinline__ void issue_async_chunk(uint32_t lds_base, const float* gsrc) {
  uint64_t ga = (uint64_t)(uintptr_t)gsrc;
  asm volatile("global_load_async_to_lds_b128 %0, %1, off"            :: "v"(lds_base), "v"(ga) : "memory");
  asm volatile("global_load_async_to_lds_b128 %0, %1, off offset:16"  :: "v"(lds_base), "v"(ga) : "memory");
  asm volatile("global_load_async_to_lds_b128 %0, %1, off offset:32"  :: "v"(lds_base), "v"(ga) : "memory");
  asm volatile("global_load_async_to_lds_b128 %0, %1, off offset:48"  :: "v"(lds_base), "v"(ga) : "memory");
  asm volatile("global_load_async_to_lds_b128 %0, %1, off offset:64"  :: "v"(lds_base), "v"(ga) : "memory");
  asm volatile("global_load_async_to_lds_b128 %0, %1, off offset:80"  :: "v"(lds_base), "v"(ga) : "memory");
  asm volatile("global_load_async_to_lds_b128 %0, %1, off offset:96"  :: "v"(lds_base), "v"(ga) : "memory");
  asm volatile("global_load_async_to_lds_b128 %0, %1, off offset:112" :: "v"(lds_base), "v"(ga) : "memory");
}

__global__ __launch_bounds__(FWD_BLK) void crf_forward(
    const float* __restrict__ observes,     // [N][C][L]
    const float* __restrict__ transitions,  // [C][C]
    const int*   __restrict__ lengths,      // [N]
    uint32_t*    __restrict__ bp_pack,      // [L][N], 8 tags x 3 bits
    int*         __restrict__ end_tag)      // [N]
{
  __shared__ float obs_lds[2 * FWD_BLK * CHUNK];   // double buffer, 8 KB

  const int tid = threadIdx.x;        // 0..31
  const int c   = tid & 7;            // this lane's "next" tag
  const int n   = blockIdx.x * SEQ_PB + (tid >> 3);

  // Per-lane contiguous observation stream: observes[n][c][t], t = 0..L-1.
  const float* stream = observes + ((size_t)n * C_TAG + c) * L_LEN;

  // Transition row in butterfly order: Tp[i] = T[c][c ^ i]  (prev = c ^ i).
  float Tp[8];
#pragma unroll
  for (int i = 0; i < 8; ++i) Tp[i] = transitions[c * C_TAG + (c ^ i)];

  const int t_end = lengths[n] - 1;

  // fv[i] == forward_var[prev = c ^ i]; all-zero init makes order irrelevant.
  float fv[8];
#pragma unroll
  for (int i = 0; i < 8; ++i) fv[i] = 0.0f;

  const uint32_t slot0 = lds_addr32(&obs_lds[tid * CHUNK]);
  const uint32_t slot1 = lds_addr32(&obs_lds[(FWD_BLK + tid) * CHUNK]);

  issue_async_chunk(slot0, stream);           // chunk 0
  issue_async_chunk(slot1, stream + CHUNK);   // chunk 1

  for (int ci = 0; ci < NCHUNK; ++ci) {
    const int buf = ci & 1;
    if (ci + 1 < NCHUNK) {
      // 16 outstanding -> wait until the other buffer's 8 remain.
      asm volatile("s_wait_asynccnt 8" ::: "memory");
    } else {
      asm volatile("s_wait_asynccnt 0" ::: "memory");
    }
    const float* lobs = &obs_lds[(buf * FWD_BLK + tid) * CHUNK];

    for (int kq = 0; kq < CHUNK / 4; ++kq) {
      float4 ov = *(const float4*)(lobs + kq * 4);   // ds_load_b128
#pragma unroll
      for (int j = 0; j < 4; ++j) {
        const int   t  = ci * CHUNK + kq * 4 + j;
        const float ob = ((const float*)&ov)[j];

        // vit_c = max_i(fv[i] + Tp[i]), argmax index in butterfly space.
        float best = fv[0] + Tp[0];
        int   bi   = 0;
#pragma unroll
        for (int i = 1; i < 8; ++i) {
          float s  = fv[i] + Tp[i];
          bool  gt = s > best;
          best = gt ? s : best;
          bi   = gt ? i : bi;
        }
        const int   bp = c ^ bi;        // actual prev tag
        const float nf = best + ob;     // new forward_var[c]

        // Pack backpointers across the 8 lanes: word |= bp << 3c, XOR-or-reduce.
        uint32_t w = (uint32_t)bp << (3 * c);
        w |= __shfl_xor(w, 1, 8);
        w |= __shfl_xor(w, 2, 8);
        w |= __shfl_xor(w, 4, 8);
        if (c == 0) bp_pack[(size_t)t * N_SEQ + n] = w;

        // Butterfly all-gather: fv[i] <- nf of lane (c ^ i).
        fv[0] = nf;
        fv[1] = __shfl_xor(fv[0], 1, 8);
        fv[2] = __shfl_xor(fv[0], 2, 8);
        fv[3] = __shfl_xor(fv[1], 2, 8);
        fv[4] = __shfl_xor(fv[0], 4, 8);
        fv[5] = __shfl_xor(fv[1], 4, 8);
        fv[6] = __shfl_xor(fv[2], 4, 8);
        fv[7] = __shfl_xor(fv[3], 4, 8);

        // end_node is only consumed at t == len-1: record it once.
        if (t == t_end) {
          // On lane 0, fv[i] = new_fv[tag i] in natural order.
          float eb = fv[0]; int ei = 0;
#pragma unroll
          for (int i = 1; i < 8; ++i) {
            bool g = fv[i] > eb;
            eb = g ? fv[i] : eb;
            ei = g ? i : ei;
          }
          if (c == 0) end_tag[n] = ei;
        }
      }
    }
    if (ci + 2 < NCHUNK) {
      issue_async_chunk(buf ? slot1 : slot0, stream + (ci + 2) * CHUNK);
    }
  }
}

__global__ __launch_bounds__(N_SEQ) void crf_backward(
    const int*      __restrict__ lengths,
    const uint32_t* __restrict__ bp_pack,   // [L][N]
    const int*      __restrict__ end_tag,   // [N]
    float*          __restrict__ out)       // [N][L]
{
  const int n   = threadIdx.x;              // one lane per sequence
  const int len = lengths[n];
  const int et  = end_tag[n];
  int best = 0;

  for (int t4 = L_LEN - 4; t4 >= 0; t4 -= 4) {
    if (t4 >= 8) {  // prefetch 2 KB ahead of the coalesced bp stream
      __builtin_prefetch(&bp_pack[(size_t)(t4 - 8) * N_SEQ + n], 0, 0);
    }
    float4 o;
#pragma unroll
    for (int j = 3; j >= 0; --j) {
      const int t = t4 + j;
      uint32_t w = bp_pack[(size_t)t * N_SEQ + n];
      if (t == len - 1) best = et;          // jump in at the chain's end
      int nb = (int)((w >> (3 * best)) & 7u);
      ((float*)&o)[j] = (t < len) ? (float)nb : -1.0f;
      best = nb;                            // follow bp BEFORE emitting (ref semantics)
    }
    *(float4*)(out + (size_t)n * L_LEN + t4) = o;
  }
}

extern "C" void kernel_launch(void* const* d_in, const int* in_sizes, int n_in,
                              void* d_out, int out_size, void* d_ws, size_t ws_size,
                              hipStream_t stream) {
  const float* observes    = (const float*)d_in[0];   // [64][8][32768] f32
  const float* transitions = (const float*)d_in[1];   // [8][8] f32
  const int*   lengths     = (const int*)d_in[2];     // [64] i32

  uint32_t* bp  = (uint32_t*)d_ws;                                   // 8 MB
  int* end_tags = (int*)((char*)d_ws + (size_t)L_LEN * N_SEQ * 4);   // 256 B

  crf_forward<<<N_SEQ / SEQ_PB, FWD_BLK, 0, stream>>>(observes, transitions,
                                                      lengths, bp, end_tags);
  crf_backward<<<1, N_SEQ, 0, stream>>>(lengths, bp, end_tags, (float*)d_out);
}